// dgl_GAT_77086073028869
// MI455X (gfx1250) — compile-verified
//
#include <hip/hip_runtime.h>
#include <math.h>

#define N_NODES 50000
#define N_EDGES 800000

typedef __attribute__((ext_vector_type(2))) float v2f;
typedef __attribute__((ext_vector_type(8))) float v8f;

// ---------------- float atomic max via sign-split integer atomics ----------
// Valid given init value of -inf. Nonneg values ordered correctly as int;
// negative values ordered reverse as uint -> use uMin.
__device__ __forceinline__ void atomicMaxF(float* addr, float val) {
    if (val >= 0.0f) {
        atomicMax((int*)addr, __float_as_int(val));
    } else {
        atomicMin((unsigned int*)addr, __float_as_uint(val));
    }
}

// ---------------- generic fill ----------------
__global__ void fillf_kernel(float* __restrict__ p, float v, int n) {
    int i = blockIdx.x * blockDim.x + threadIdx.x;
    if (i < n) p[i] = v;
}

// ---------------- WMMA f32 GEMM: feat = H[N,DIN] @ W[DIN,DOUT] -------------
// Strip-mined: one wave computes MT consecutive 16x16 M-tiles for one N-tile.
// B fragment loaded once per k-step and reused by MT independent WMMA chains.
// N_NODES/16 = 3125 = 5^5, so MT=5 divides exactly (625 strips).
// A-frag (16x4 f32): lane l -> M = l%16, K = 2*(l/16)+v  (v in {0,1})
// B-frag (4x16 f32): lane l -> N = l%16, K = 2*(l/16)+v
// C/D  (16x16 f32): lane l, vgpr v -> M = v + 8*(l/16), N = l%16
template <int DIN, int DOUT, int MT>
__global__ __launch_bounds__(128) void gemm_wmma_f32(
    const float* __restrict__ Hin, const float* __restrict__ W,
    float* __restrict__ Fout) {
    const int tiles_n = DOUT / 16;
    const int strips  = (N_NODES / 16) / MT;   // 625
    const int total   = strips * tiles_n;
    int wave = blockIdx.x * (blockDim.x >> 5) + (threadIdx.x >> 5);
    if (wave >= total) return;  // wave-uniform: EXEC stays all-1s for WMMA
    int ts = wave / tiles_n;
    int tn = wave % tiles_n;

    int lane = threadIdx.x & 31;
    int half = lane >> 4;   // 0 or 1
    int mr   = lane & 15;   // M (for A) or N (for B/C/D)

    const float* abase = Hin + (size_t)(ts * MT * 16 + mr) * DIN + 2 * half;
    const float* bptr  = W + tn * 16 + mr;

    v8f c[MT];
    #pragma unroll
    for (int m = 0; m < MT; ++m) c[m] = (v8f){};

    #pragma unroll 2
    for (int k = 0; k < DIN; k += 4) {
        v2f b;
        b.x = bptr[(size_t)(k + 2 * half + 0) * DOUT];
        b.y = bptr[(size_t)(k + 2 * half + 1) * DOUT];
        #pragma unroll
        for (int m = 0; m < MT; ++m) {
            const float* ap = abase + (size_t)m * 16 * DIN;
            v2f a;
            a.x = ap[k + 0];
            a.y = ap[k + 1];
            c[m] = __builtin_amdgcn_wmma_f32_16x16x4_f32(
                       /*neg_a=*/false, a, /*neg_b=*/false, b,
                       /*c_mod=*/(short)0, c[m],
                       /*reuse_a=*/false, /*reuse_b=*/false);
        }
    }

    #pragma unroll
    for (int m = 0; m < MT; ++m) {
        float* out = Fout + (size_t)((ts * MT + m) * 16 + 8 * half) * DOUT +
                     tn * 16 + mr;
        #pragma unroll
        for (int v = 0; v < 8; ++v) out[(size_t)v * DOUT] = c[m][v];
    }
}

// ---------------- per-node attention projections el/er ---------------------
template <int H, int F>
__global__ void attn_proj(const float* __restrict__ feat,
                          const float* __restrict__ al,
                          const float* __restrict__ ar,
                          float* __restrict__ el, float* __restrict__ er) {
    int t = blockIdx.x * blockDim.x + threadIdx.x;
    if (t >= N_NODES * H) return;
    int n = t / H, h = t % H;
    const float* f = feat + (size_t)n * (H * F) + h * F;
    float sl = 0.f, sr = 0.f;
    #pragma unroll
    for (int i = 0; i < F; ++i) {
        float fv = f[i];
        sl += fv * al[h * F + i];
        sr += fv * ar[h * F + i];
    }
    el[t] = sl;
    er[t] = sr;
}

// ---------------- edge pass 1: leaky-relu score + segment max --------------
template <int H>
__global__ void edge_max(const int* __restrict__ src, const int* __restrict__ dst,
                         const float* __restrict__ el, const float* __restrict__ er,
                         float* __restrict__ ebuf, float* __restrict__ m) {
    int t = blockIdx.x * blockDim.x + threadIdx.x;
    if (t >= N_EDGES * H) return;
    int e = t / H, h = t % H;
    int s = src[e], d = dst[e];
    float v = el[s * H + h] + er[d * H + h];
    v = (v > 0.f) ? v : 0.2f * v;  // LeakyReLU(0.2)
    ebuf[t] = v;
    atomicMaxF(&m[d * H + h], v);
}

// ---------------- edge pass 2: exp + segment sum ---------------------------
template <int H>
__global__ void edge_exp(const int* __restrict__ dst,
                         const float* __restrict__ m,
                         float* __restrict__ ebuf, float* __restrict__ ssum) {
    int t = blockIdx.x * blockDim.x + threadIdx.x;
    if (t >= N_EDGES * H) return;
    int e = t / H, h = t % H;
    int d = dst[e];
    float ex = __expf(ebuf[t] - m[d * H + h]);
    ebuf[t] = ex;
    atomicAdd(&ssum[d * H + h], ex);
}

// ---------------- edge pass 3: alpha-weighted scatter aggregation ----------
// One wave per edge; each lane handles H*F/32 consecutive features, which all
// belong to a single head (F % VPL == 0), so alpha is computed once per lane.
template <int H, int F>
__global__ __launch_bounds__(256) void edge_aggr(
    const int* __restrict__ src, const int* __restrict__ dst,
    const float* __restrict__ feat, const float* __restrict__ ebuf,
    const float* __restrict__ ssum, float* __restrict__ acc) {
    const int D   = H * F;
    const int VPL = D / 32;  // 4 (layers 0/1), 2 (layer 2)
    static_assert(F % VPL == 0, "lane features must stay within one head");
    int wave = blockIdx.x * (blockDim.x >> 5) + (threadIdx.x >> 5);
    if (wave >= N_EDGES) return;
    int lane = threadIdx.x & 31;
    int s = src[wave], d = dst[wave];
    int base = lane * VPL;
    int h = base / F;  // constant across the VPL features of this lane
    float alpha = ebuf[(size_t)wave * H + h] / ssum[d * H + h];
    const float* fp = feat + (size_t)s * D + base;
    float* ap = acc + (size_t)d * D + base;
    #pragma unroll
    for (int j = 0; j < VPL; ++j) {
        atomicAdd(&ap[j], fp[j] * alpha);
    }
}

// ---------------- epilogues ------------------------------------------------
__global__ void epilogue_elu(const float* __restrict__ acc,
                             const float* __restrict__ bias,
                             float* __restrict__ hout, int total, int D) {
    int t = blockIdx.x * blockDim.x + threadIdx.x;
    if (t >= total) return;
    float v = acc[t] + bias[t % D];
    hout[t] = (v > 0.f) ? v : (__expf(v) - 1.f);  // ELU(alpha=1)
}

__global__ void epilogue_final(const float* __restrict__ acc,
                               const float* __restrict__ bias,
                               float* __restrict__ out, int total, int D) {
    int t = blockIdx.x * blockDim.x + threadIdx.x;
    if (t >= total) return;
    out[t] = acc[t] + bias[t % D];  // H==1 -> mean over heads is identity
}

// ---------------- host-side per-layer driver -------------------------------
template <int H, int F>
static void run_attention(const int* src, const int* dst, const float* feat,
                          const float* al, const float* ar,
                          float* el, float* er, float* mbuf, float* sbuf,
                          float* ebuf, float* acc, hipStream_t stream) {
    const int NH = N_NODES * H;
    const int ND = N_NODES * H * F;
    const int EH = N_EDGES * H;
    attn_proj<H, F><<<(NH + 255) / 256, 256, 0, stream>>>(feat, al, ar, el, er);
    fillf_kernel<<<(NH + 255) / 256, 256, 0, stream>>>(mbuf, -__builtin_huge_valf(), NH);
    fillf_kernel<<<(NH + 255) / 256, 256, 0, stream>>>(sbuf, 0.f, NH);
    fillf_kernel<<<(ND + 255) / 256, 256, 0, stream>>>(acc, 0.f, ND);
    edge_max<H><<<(EH + 255) / 256, 256, 0, stream>>>(src, dst, el, er, ebuf, mbuf);
    edge_exp<H><<<(EH + 255) / 256, 256, 0, stream>>>(dst, mbuf, ebuf, sbuf);
    edge_aggr<H, F><<<(N_EDGES + 7) / 8, 256, 0, stream>>>(src, dst, feat, ebuf, sbuf, acc);
}

template <int DIN, int DOUT>
static void launch_gemm(const float* h, const float* W, float* f, hipStream_t stream) {
    const int MT = 5;  // 3125 M-tiles = 625 strips of 5
    const int total = ((N_NODES / 16) / MT) * (DOUT / 16);
    const int wpb = 4;  // 4 waves (128 threads) per block
    gemm_wmma_f32<DIN, DOUT, MT>
        <<<(total + wpb - 1) / wpb, wpb * 32, 0, stream>>>(h, W, f);
}

extern "C" void kernel_launch(void* const* d_in, const int* in_sizes, int n_in,
                              void* d_out, int out_size, void* d_ws, size_t ws_size,
                              hipStream_t stream) {
    const float* x   = (const float*)d_in[0];
    const int*   src = (const int*)d_in[1];
    const int*   dst = (const int*)d_in[2];
    const float* W0  = (const float*)d_in[3];
    const float* al0 = (const float*)d_in[4];
    const float* ar0 = (const float*)d_in[5];
    const float* b0  = (const float*)d_in[6];
    const float* W1  = (const float*)d_in[7];
    const float* al1 = (const float*)d_in[8];
    const float* ar1 = (const float*)d_in[9];
    const float* b1  = (const float*)d_in[10];
    const float* W2  = (const float*)d_in[11];
    const float* al2 = (const float*)d_in[12];
    const float* ar2 = (const float*)d_in[13];
    const float* b2  = (const float*)d_in[14];
    float* out = (float*)d_out;

    // workspace carve-up (floats)
    float* ws = (float*)d_ws;
    size_t off = 0;
    float* feat = ws + off; off += (size_t)N_NODES * 128;
    float* hbuf = ws + off; off += (size_t)N_NODES * 128;
    float* acc  = ws + off; off += (size_t)N_NODES * 128;
    float* el   = ws + off; off += (size_t)N_NODES * 8;
    float* er   = ws + off; off += (size_t)N_NODES * 8;
    float* mbuf = ws + off; off += (size_t)N_NODES * 8;
    float* sbuf = ws + off; off += (size_t)N_NODES * 8;
    float* ebuf = ws + off; off += (size_t)N_EDGES * 8;
    (void)ws_size; (void)n_in; (void)in_sizes; (void)out_size;

    // ---- Layer 0: 256 -> 8 heads x 16, ELU ----
    launch_gemm<256, 128>(x, W0, feat, stream);
    run_attention<8, 16>(src, dst, feat, al0, ar0, el, er, mbuf, sbuf, ebuf, acc, stream);
    epilogue_elu<<<(N_NODES * 128 + 255) / 256, 256, 0, stream>>>(acc, b0, hbuf, N_NODES * 128, 128);

    // ---- Layer 1: 128 -> 8 heads x 16, ELU ----
    launch_gemm<128, 128>(hbuf, W1, feat, stream);
    run_attention<8, 16>(src, dst, feat, al1, ar1, el, er, mbuf, sbuf, ebuf, acc, stream);
    epilogue_elu<<<(N_NODES * 128 + 255) / 256, 256, 0, stream>>>(acc, b1, hbuf, N_NODES * 128, 128);

    // ---- Layer 2: 128 -> 1 head x 64, mean over heads (identity) ----
    launch_gemm<128, 64>(hbuf, W2, feat, stream);
    run_attention<1, 64>(src, dst, feat, al2, ar2, el, er, mbuf, sbuf, ebuf, acc, stream);
    epilogue_final<<<(N_NODES * 64 + 255) / 256, 256, 0, stream>>>(acc, b2, out, N_NODES * 64, 64);
}